// MessageBlock_13005160972668
// MI455X (gfx1250) — compile-verified
//
#include <hip/hip_runtime.h>
#include <math.h>

#define FEAT   128
#define F3     (3 * FEAT)
#define NRBF   20
#define KDIM   128
#define CUTOFF 5.0f
#define PI_F   3.14159265358979323846f

typedef float v2f __attribute__((ext_vector_type(2)));
typedef float v8f __attribute__((ext_vector_type(8)));

__global__ void zero_kernel(float* __restrict__ p, int n) {
    int i = blockIdx.x * blockDim.x + threadIdx.x;
    if (i < n) p[i] = 0.0f;
}

// C[M,N] = act(A[M,K=128] @ B[128,N] + bias[N]) ; act = SiLU if APPLY_SILU.
// One wave32 per 16x16 output tile, V_WMMA_F32_16X16X4_F32, K unrolled 32x.
// N compile-time so all B loads use immediate offsets (no per-step addr math).
template <int N, int APPLY_SILU>
__global__ __launch_bounds__(256) void gemm_bias_act_wmma(
    const float* __restrict__ A, const float* __restrict__ B,
    const float* __restrict__ bias, float* __restrict__ C,
    int totalTiles)
{
    const int lane = threadIdx.x & 31;
    const int wave = threadIdx.x >> 5;
    const int nTilesN = N >> 4;
    const int t = blockIdx.x * 8 + wave;
    if (t >= totalTiles) return;              // wave-uniform: EXEC stays all-1s
    const int mTile = t / nTilesN;
    const int nTile = t % nTilesN;
    const int l15   = lane & 15;
    const int kHalf = (lane >> 4) << 1;       // 0 or 2
    const int nCol  = nTile * 16 + l15;

    const float* aPtr = A + (size_t)(mTile * 16 + l15) * KDIM + kHalf;
    const float* bPtr = B + (size_t)kHalf * N + nCol;

    v8f acc = {};
#pragma unroll
    for (int kb = 0; kb < KDIM / 4; ++kb) {
        v2f a, b;
        a[0] = aPtr[kb * 4 + 0];              // A[m][k], A[m][k+1] (contiguous)
        a[1] = aPtr[kb * 4 + 1];
        b[0] = bPtr[kb * 4 * N];              // B[k][n]   (immediate offsets)
        b[1] = bPtr[(kb * 4 + 1) * N];        // B[k+1][n]
        acc = __builtin_amdgcn_wmma_f32_16x16x4_f32(
            /*neg_a=*/false, a, /*neg_b=*/false, b,
            /*c_mod=*/(short)0, acc, /*reuse_a=*/false, /*reuse_b=*/false);
    }

    const float bv = bias[nCol];
    const int rowBase = mTile * 16 + (lane >> 4) * 8;
#pragma unroll
    for (int r = 0; r < 8; ++r) {
        float x = acc[r] + bv;
        if (APPLY_SILU) x = x * (1.0f / (1.0f + expf(-x)));
        C[(size_t)(rowBase + r) * N + nCol] = x;
    }
}

// Fused edge kernel: 128 lanes per edge, 2 edges in flight per 256-thread block.
// Wd/bd cached in LDS; per-edge rbf*env double-buffered through LDS (1 barrier
// per iteration); scatter via hardware f32 atomics.
__global__ __launch_bounds__(256) void edge_message_kernel(
    const float* __restrict__ phi,        // [nNodes, 3*FEAT]
    const float* __restrict__ v_j,        // [nNodes, FEAT, 3]
    const float* __restrict__ r_ij,       // [E, 3]
    const long long* __restrict__ nbrs,   // [E, 2] int64
    const float* __restrict__ Wd,         // [NRBF, 3*FEAT]
    const float* __restrict__ bd,         // [3*FEAT]
    float* __restrict__ ds_out,           // [nNodes, FEAT]
    float* __restrict__ dv_out,           // [nNodes, FEAT, 3]
    int edgesPerGroup)
{
    __shared__ float sWd[NRBF * F3];
    __shared__ float sBd[F3];
    __shared__ float sRe[2][2][NRBF];     // [group][buffer][k]

    // Vectorized LDS staging of Wd (7680 floats) and bd (384 floats).
    {
        const float4* Wd4 = (const float4*)Wd;
        float4* sWd4 = (float4*)sWd;
        for (int idx = threadIdx.x; idx < (NRBF * F3) / 4; idx += blockDim.x)
            sWd4[idx] = Wd4[idx];
        for (int idx = threadIdx.x; idx < F3; idx += blockDim.x)
            sBd[idx] = bd[idx];
    }
    __syncthreads();

    const int group = threadIdx.x >> 7;       // 0 or 1
    const int lane  = threadIdx.x & 127;
    const long long eBase = (long long)(blockIdx.x * 2 + group) * edgesPerGroup;

    for (int it = 0; it < edgesPerGroup; ++it) {
        const long long e = eBase + it;
        const int buf = it & 1;
        const long long i = nbrs[2 * e + 0];
        const long long j = nbrs[2 * e + 1];

        const float rx = r_ij[3 * e + 0];
        const float ry = r_ij[3 * e + 1];
        const float rz = r_ij[3 * e + 2];
        const float d2   = rx * rx + ry * ry + rz * rz + 3.0f * 1e-15f;
        const float dist = sqrtf(d2);
        const float rinv = 1.0f / dist;
        const float ux = rx * rinv, uy = ry * rinv, uz = rz * rinv;
        const float env = (dist < CUTOFF)
                            ? 0.5f * (__cosf(PI_F * dist * (1.0f / CUTOFF)) + 1.0f)
                            : 0.0f;

        if (lane < NRBF) {
            const float kf = (float)(lane + 1) * (PI_F / CUTOFF);
            sRe[group][buf][lane] = __sinf(dist * kf) * rinv * env;   // rbf * env
        }
        __syncthreads();

        float w0 = sBd[lane] * env;
        float w1 = sBd[lane + FEAT] * env;
        float w2 = sBd[lane + 2 * FEAT] * env;
#pragma unroll
        for (int k = 0; k < NRBF; ++k) {
            const float re = sRe[group][buf][k];
            const float* wrow = &sWd[k * F3 + lane];
            w0 = fmaf(re, wrow[0], w0);
            w1 = fmaf(re, wrow[FEAT], w1);
            w2 = fmaf(re, wrow[2 * FEAT], w2);
        }

        const float* ph = phi + (size_t)j * F3;
        const float inv0 = ph[lane] * w0;
        const float inv1 = ph[lane + FEAT] * w1;
        const float inv2 = ph[lane + 2 * FEAT] * w2;

        const float* vj = v_j + ((size_t)j * FEAT + lane) * 3;
        const float dv0 = fmaf(inv0, vj[0], inv2 * ux);
        const float dv1 = fmaf(inv0, vj[1], inv2 * uy);
        const float dv2 = fmaf(inv0, vj[2], inv2 * uz);

        unsafeAtomicAdd(&ds_out[(size_t)i * FEAT + lane], inv1);
        float* dvp = dv_out + ((size_t)i * FEAT + lane) * 3;
        unsafeAtomicAdd(&dvp[0], dv0);
        unsafeAtomicAdd(&dvp[1], dv1);
        unsafeAtomicAdd(&dvp[2], dv2);
        // no trailing barrier: sRe is double-buffered
    }
}

extern "C" void kernel_launch(void* const* d_in, const int* in_sizes, int n_in,
                              void* d_out, int out_size, void* d_ws, size_t ws_size,
                              hipStream_t stream) {
    const float*     s_j  = (const float*)d_in[0];
    const float*     v_j  = (const float*)d_in[1];
    const float*     r_ij = (const float*)d_in[2];
    const long long* nbrs = (const long long*)d_in[3];
    const float*     W1   = (const float*)d_in[4];
    const float*     b1   = (const float*)d_in[5];
    const float*     W2   = (const float*)d_in[6];
    const float*     b2   = (const float*)d_in[7];
    const float*     Wd   = (const float*)d_in[8];
    const float*     bd   = (const float*)d_in[9];

    const int nNodes = in_sizes[0] / FEAT;    // 20000
    const int nEdges = in_sizes[2] / 3;       // 640000

    float* H      = (float*)d_ws;                       // [nNodes, FEAT]
    float* phi    = H + (size_t)nNodes * FEAT;          // [nNodes, 3*FEAT]
    float* ds_out = (float*)d_out;                      // [nNodes, FEAT]
    float* dv_out = ds_out + (size_t)nNodes * FEAT;     // [nNodes, FEAT, 3]

    // 1) zero accumulation targets (harness poisons d_out; atomics accumulate)
    zero_kernel<<<(out_size + 255) / 256, 256, 0, stream>>>((float*)d_out, out_size);

    // 2) H = silu(s_j @ W1 + b1)
    const int t1 = (nNodes / 16) * (FEAT / 16);
    gemm_bias_act_wmma<FEAT, 1><<<(t1 + 7) / 8, 256, 0, stream>>>(s_j, W1, b1, H, t1);

    // 3) phi = H @ W2 + b2
    const int t2 = (nNodes / 16) * (F3 / 16);
    gemm_bias_act_wmma<F3, 0><<<(t2 + 7) / 8, 256, 0, stream>>>(H, W2, b2, phi, t2);

    // 4) fused edge gather / RBF projection / scatter-add
    const int edgesPerGroup = 64;
    const int blocks = nEdges / (2 * edgesPerGroup);    // 5000
    edge_message_kernel<<<blocks, 256, 0, stream>>>(phi, v_j, r_ij, nbrs, Wd, bd,
                                                    ds_out, dv_out, edgesPerGroup);
}